// NetGINE_68006512165297
// MI455X (gfx1250) — compile-verified
//
#include <hip/hip_runtime.h>
#include <hip/hip_bf16.h>

// ---------------------------------------------------------------------------
// NetGINE on MI455X (gfx1250): fused WMMA-FP32 GIN-E layers.
//  - bond encoder (2 GEMMs) + message + scatter fused per 16-edge strip
//  - node MLP (2 GEMMs) fused per 16-node strip
//  - LDS-staged 16x256 intermediates (stride-260 padding, bank-conflict free)
//  - V_WMMA_F32_16X16X4_F32 for all matmuls (fp32 end-to-end)
// ---------------------------------------------------------------------------

typedef __attribute__((ext_vector_type(2))) float v2f;
typedef __attribute__((ext_vector_type(8))) float v8f;
typedef __attribute__((ext_vector_type(4))) float v4f;

#define GN 50000
#define GE 300000
#define GFE 64
#define GH 256
#define GG 256
#define GL 4
#define GOUT 128
#define LHW (GL * GH)   // 1024
#define TSTR 260        // LDS row stride (words), pads 256 -> 16 distinct banks

__device__ __forceinline__ v8f wmma4(v2f a, v2f b, v8f c) {
  // D = A(16x4 f32) x B(4x16 f32) + C(16x16 f32)
  return __builtin_amdgcn_wmma_f32_16x16x4_f32(
      /*neg_a=*/false, a, /*neg_b=*/false, b,
      /*c_mod=*/(short)0, c, /*reuse_a=*/false, /*reuse_b=*/false);
}

__device__ __forceinline__ float relu_f(float v) { return v > 0.f ? v : 0.f; }

// --------------------------------------------------------------------------
// aggr = (1 + eps[l]) * x     (fully overwrites aggr)
// --------------------------------------------------------------------------
__global__ void init_aggr_kernel(float* __restrict__ aggr,
                                 const float* __restrict__ x, int ldx,
                                 const float* __restrict__ eps, int l) {
  size_t i = (size_t)blockIdx.x * blockDim.x + threadIdx.x;
  if (i >= (size_t)GN * GH) return;
  int n = (int)(i >> 8);
  int c = (int)(i & 255);
  float s = 1.f + eps[l];
  aggr[i] = s * x[(size_t)n * ldx + c];
}

// --------------------------------------------------------------------------
// Fused bond encoder + message + scatter.
// Per wave: one 16-edge strip.
//   Phase 1: T = relu(EA[strip] @ W1 + b1)           -> LDS  [16 x 256]
//   Phase 2: Etile = T @ W2 + b2 (per 16x16 n-tile)
//            msg = relu(x[src] + Etile); atomicAdd into aggr[dst]
// --------------------------------------------------------------------------
__global__ void bond_msg_kernel(const float* __restrict__ ea,
                                const int* __restrict__ src,
                                const int* __restrict__ dst,
                                const float* __restrict__ x, int ldx,
                                const float* __restrict__ W1,
                                const float* __restrict__ b1,
                                const float* __restrict__ W2,
                                const float* __restrict__ b2,
                                float* __restrict__ aggr, int nstrips) {
  extern __shared__ float lds[];
  const int wave = threadIdx.x >> 5;
  const int lane = threadIdx.x & 31;
  float* t = lds + (size_t)wave * (16 * TSTR);

  const int strip = blockIdx.x * (blockDim.x >> 5) + wave;
  if (strip >= nstrips) return;
  const int e0 = strip * 16;

  const int m  = lane & 15;        // A row / B col / C col within tile
  const int kh = (lane >> 4) * 2;  // K sub-offset of this half-wave
  const int rb = (lane >> 4) * 8;  // C row base of this half-wave

  // ---- Phase 1: T = relu(EA @ W1 + b1) ----
  for (int nt = 0; nt < GH / 16; ++nt) {
    const int n0 = nt * 16;
    v8f c = {};
    for (int k = 0; k < GFE; k += 4) {
      v2f a, b;
      const float* ap = ea + (size_t)(e0 + m) * GFE + k + kh;
      a.x = ap[0]; a.y = ap[1];
      const float* bp = W1 + (size_t)(k + kh) * GH + n0 + m;
      b.x = bp[0]; b.y = bp[GH];
      c = wmma4(a, b, c);
    }
    const int n = n0 + m;
    const float bias = b1[n];
#pragma unroll
    for (int r = 0; r < 8; ++r)
      t[(rb + r) * TSTR + n] = relu_f(c[r] + bias);
  }

  // order phase-1 LDS stores before phase-2 cross-lane LDS reads
  asm volatile("s_wait_dscnt 0" ::: "memory");

  // ---- Phase 2: E = T @ W2 + b2 ; msg = relu(x[src]+E) ; scatter-add ----
  for (int nt = 0; nt < GH / 16; ++nt) {
    const int n0 = nt * 16;
    v8f c = {};
    for (int k = 0; k < GH; k += 4) {
      v2f a, b;
      const float* ap = t + m * TSTR + k + kh;
      a.x = ap[0]; a.y = ap[1];
      const float* bp = W2 + (size_t)(k + kh) * GH + n0 + m;
      b.x = bp[0]; b.y = bp[GH];
      c = wmma4(a, b, c);
    }
    const int n = n0 + m;
    const float bias = b2[n];
#pragma unroll
    for (int r = 0; r < 8; ++r) {
      const int er = e0 + rb + r;
      const int s = src[er];
      const int d = dst[er];
      const float xv = x[(size_t)s * ldx + n];
      const float v = relu_f(xv + c[r] + bias);
      atomicAdd(&aggr[(size_t)d * GH + n], v);
    }
  }
}

// --------------------------------------------------------------------------
// Fused node MLP: h2 = relu(relu(h @ W1 + b1) @ W2 + b2)
// Per wave: one 16-node strip, intermediate staged in LDS.
// --------------------------------------------------------------------------
__global__ void mlp_fused_kernel(const float* __restrict__ h,
                                 const float* __restrict__ W1,
                                 const float* __restrict__ b1,
                                 const float* __restrict__ W2,
                                 const float* __restrict__ b2,
                                 float* __restrict__ h2, int nstrips) {
  extern __shared__ float lds[];
  const int wave = threadIdx.x >> 5;
  const int lane = threadIdx.x & 31;
  float* t = lds + (size_t)wave * (16 * TSTR);

  const int strip = blockIdx.x * (blockDim.x >> 5) + wave;
  if (strip >= nstrips) return;
  const int r0 = strip * 16;

  const int m  = lane & 15;
  const int kh = (lane >> 4) * 2;
  const int rb = (lane >> 4) * 8;

  // ---- Phase 1: T = relu(h @ W1 + b1) ----
  for (int nt = 0; nt < GH / 16; ++nt) {
    const int n0 = nt * 16;
    v8f c = {};
    for (int k = 0; k < GH; k += 4) {
      v2f a, b;
      const float* ap = h + (size_t)(r0 + m) * GH + k + kh;
      a.x = ap[0]; a.y = ap[1];
      const float* bp = W1 + (size_t)(k + kh) * GH + n0 + m;
      b.x = bp[0]; b.y = bp[GH];
      c = wmma4(a, b, c);
    }
    const int n = n0 + m;
    const float bias = b1[n];
#pragma unroll
    for (int r = 0; r < 8; ++r)
      t[(rb + r) * TSTR + n] = relu_f(c[r] + bias);
  }

  asm volatile("s_wait_dscnt 0" ::: "memory");

  // ---- Phase 2: h2 = relu(T @ W2 + b2) ----
  for (int nt = 0; nt < GH / 16; ++nt) {
    const int n0 = nt * 16;
    v8f c = {};
    for (int k = 0; k < GH; k += 4) {
      v2f a, b;
      const float* ap = t + m * TSTR + k + kh;
      a.x = ap[0]; a.y = ap[1];
      const float* bp = W2 + (size_t)(k + kh) * GH + n0 + m;
      b.x = bp[0]; b.y = bp[GH];
      c = wmma4(a, b, c);
    }
    const int n = n0 + m;
    const float bias = b2[n];
#pragma unroll
    for (int r = 0; r < 8; ++r)
      h2[(size_t)(r0 + rb + r) * GH + n] = relu_f(c[r] + bias);
  }
}

// --------------------------------------------------------------------------
// Generic WMMA GEMM: C = [relu](A[M,K] @ W[K,Nc] + bias)
// One 16x16 output tile per wave.
// --------------------------------------------------------------------------
__global__ void wmma_gemm_kernel(const float* __restrict__ A, int lda,
                                 const float* __restrict__ W, int ldw,
                                 const float* __restrict__ bias,
                                 float* __restrict__ C, int ldc,
                                 int M, int Nc, int K, int do_relu) {
  const int wave = threadIdx.x >> 5;
  const int lane = threadIdx.x & 31;
  const int gw = blockIdx.x * (blockDim.x >> 5) + wave;
  const int ntiles = Nc / 16;
  const int mt = gw / ntiles;
  const int nt = gw % ntiles;
  if (mt * 16 >= M) return;

  const int m  = lane & 15;
  const int kh = (lane >> 4) * 2;
  const int rb = (lane >> 4) * 8;
  const int n0 = nt * 16;

  v8f c = {};
  for (int k = 0; k < K; k += 4) {
    v2f a, b;
    const float* ap = A + (size_t)(mt * 16 + m) * lda + k + kh;
    a.x = ap[0]; a.y = ap[1];
    const float* bp = W + (size_t)(k + kh) * ldw + n0 + m;
    b.x = bp[0]; b.y = bp[ldw];
    c = wmma4(a, b, c);
  }
  const int n = n0 + m;
  const float bb = bias ? bias[n] : 0.f;
#pragma unroll
  for (int r = 0; r < 8; ++r) {
    float v = c[r] + bb;
    if (do_relu) v = relu_f(v);
    C[(size_t)(mt * 16 + rb + r) * ldc + n] = v;
  }
}

// --------------------------------------------------------------------------
// BatchNorm statistics: per-channel sum and sum-of-squares over N rows.
// Block = 256 threads (one per channel), processes 256 rows.
// --------------------------------------------------------------------------
__global__ void bn_stats_kernel(const float* __restrict__ h2,
                                float* __restrict__ stats) {
  const int c = threadIdx.x;
  const int row0 = blockIdx.x * 256;
  int rend = row0 + 256;
  if (rend > GN) rend = GN;
  float s = 0.f, q = 0.f;
  for (int n = row0; n < rend; ++n) {
    float v = h2[(size_t)n * GH + c];
    s += v;
    q += v * v;
  }
  atomicAdd(&stats[c], s);
  atomicAdd(&stats[GH + c], q);
}

__global__ void bn_apply_kernel(const float* __restrict__ h2,
                                const float* __restrict__ stats,
                                const float* __restrict__ gamma,
                                const float* __restrict__ beta,
                                float* __restrict__ xc_col) {
  size_t i = (size_t)blockIdx.x * blockDim.x + threadIdx.x;
  if (i >= (size_t)GN * GH) return;
  const int n = (int)(i >> 8);
  const int c = (int)(i & 255);
  const float inv_n = 1.f / (float)GN;
  const float mu = stats[c] * inv_n;
  float var = stats[GH + c] * inv_n - mu * mu;
  const float inv = rsqrtf(var + 1e-5f);
  xc_col[(size_t)n * LHW + c] = (h2[i] - mu) * inv * gamma[c] + beta[c];
}

// --------------------------------------------------------------------------
// Pooling helpers
// --------------------------------------------------------------------------
__global__ void zero_kernel(float* __restrict__ p, size_t count) {
  size_t i = (size_t)blockIdx.x * blockDim.x + threadIdx.x;
  if (i < count) p[i] = 0.f;
}

__global__ void pool_accum_kernel(const float* __restrict__ xc,
                                  const int* __restrict__ batch,
                                  float* __restrict__ pooled) {
  size_t i = (size_t)blockIdx.x * blockDim.x + threadIdx.x;
  if (i >= (size_t)GN * LHW) return;
  const int n = (int)(i >> 10);
  const int c = (int)(i & 1023);
  const int g = batch[n];
  atomicAdd(&pooled[(size_t)g * LHW + c], xc[i]);
}

__global__ void count_kernel(const int* __restrict__ batch,
                             float* __restrict__ cnts) {
  int n = blockIdx.x * blockDim.x + threadIdx.x;
  if (n >= GN) return;
  atomicAdd(&cnts[batch[n]], 1.f);
}

__global__ void pool_norm_kernel(float* __restrict__ pooled,
                                 const float* __restrict__ cnts) {
  size_t i = (size_t)blockIdx.x * blockDim.x + threadIdx.x;
  if (i >= (size_t)GG * LHW) return;
  const int g = (int)(i >> 10);
  pooled[i] /= fmaxf(cnts[g], 1.f);
}

// --------------------------------------------------------------------------
// Host launcher
// --------------------------------------------------------------------------
extern "C" void kernel_launch(void* const* d_in, const int* in_sizes, int n_in,
                              void* d_out, int out_size, void* d_ws, size_t ws_size,
                              hipStream_t stream) {
  const float* x        = (const float*)d_in[0];
  const int*   ei       = (const int*)d_in[1];
  const float* ea       = (const float*)d_in[2];
  const int*   batch    = (const int*)d_in[3];
  const float* bond_W1  = (const float*)d_in[4];
  const float* bond_b1  = (const float*)d_in[5];
  const float* bond_W2  = (const float*)d_in[6];
  const float* bond_b2  = (const float*)d_in[7];
  const float* mlp_W1   = (const float*)d_in[8];
  const float* mlp_b1   = (const float*)d_in[9];
  const float* mlp_W2   = (const float*)d_in[10];
  const float* mlp_b2   = (const float*)d_in[11];
  const float* eps      = (const float*)d_in[12];
  const float* bn_gamma = (const float*)d_in[13];
  const float* bn_beta  = (const float*)d_in[14];
  const float* fc1_W    = (const float*)d_in[15];
  const float* fc1_b    = (const float*)d_in[16];
  const float* fc4_W    = (const float*)d_in[17];
  const float* fc4_b    = (const float*)d_in[18];
  const int* src = ei;
  const int* dst = ei + GE;

  // workspace carve-up (floats)
  float* ws     = (float*)d_ws;
  float* aggr   = ws;                                  // N*H
  float* h2     = aggr + (size_t)GN * GH;              // N*H
  float* xc     = h2 + (size_t)GN * GH;                // N*L*H
  float* stats  = xc + (size_t)GN * LHW;               // 2*H
  float* pooled = stats + 2 * GH;                      // G*L*H
  float* cnts   = pooled + (size_t)GG * LHW;           // G
  float* fcb    = cnts + GG;                           // G*H

  const int stripsE = GE / 16;                         // 18750
  const int stripsN = GN / 16;                         // 3125
  const int wavesPerBlk = 8;
  const int blocksE = (stripsE + wavesPerBlk - 1) / wavesPerBlk;
  const int blocksN = (stripsN + wavesPerBlk - 1) / wavesPerBlk;
  const size_t ldsBytes = (size_t)wavesPerBlk * 16 * TSTR * sizeof(float); // 133120

  const int elemBlocksNH = (int)(((size_t)GN * GH + 255) / 256);   // 50000

  const float* x_cur = x;
  int ldx = GH;
  for (int l = 0; l < GL; ++l) {
    init_aggr_kernel<<<elemBlocksNH, 256, 0, stream>>>(aggr, x_cur, ldx, eps, l);
    bond_msg_kernel<<<blocksE, 32 * wavesPerBlk, ldsBytes, stream>>>(
        ea, src, dst, x_cur, ldx,
        bond_W1 + (size_t)l * GFE * GH, bond_b1 + (size_t)l * GH,
        bond_W2 + (size_t)l * GH * GH, bond_b2 + (size_t)l * GH,
        aggr, stripsE);
    mlp_fused_kernel<<<blocksN, 32 * wavesPerBlk, ldsBytes, stream>>>(
        aggr,
        mlp_W1 + (size_t)l * GH * GH, mlp_b1 + (size_t)l * GH,
        mlp_W2 + (size_t)l * GH * GH, mlp_b2 + (size_t)l * GH,
        h2, stripsN);
    zero_kernel<<<2, 256, 0, stream>>>(stats, 2 * GH);
    bn_stats_kernel<<<(GN + 255) / 256, 256, 0, stream>>>(h2, stats);
    bn_apply_kernel<<<elemBlocksNH, 256, 0, stream>>>(
        h2, stats, bn_gamma + (size_t)l * GH, bn_beta + (size_t)l * GH,
        xc + (size_t)l * GH);
    x_cur = xc + (size_t)l * GH;
    ldx = LHW;
  }

  // pooled (+cnts contiguous) zero, accumulate, normalize
  zero_kernel<<<(int)(((size_t)GG * LHW + GG + 255) / 256), 256, 0, stream>>>(
      pooled, (size_t)GG * LHW + GG);
  pool_accum_kernel<<<(int)(((size_t)GN * LHW + 255) / 256), 256, 0, stream>>>(
      xc, batch, pooled);
  count_kernel<<<(GN + 255) / 256, 256, 0, stream>>>(batch, cnts);
  pool_norm_kernel<<<(int)(((size_t)GG * LHW + 255) / 256), 256, 0, stream>>>(
      pooled, cnts);

  // fc1: [G,1024] @ [1024,256] + b, relu
  {
    const int tiles = (GG / 16) * (GH / 16);     // 256 waves
    wmma_gemm_kernel<<<tiles / 4, 128, 0, stream>>>(
        pooled, LHW, fc1_W, GH, fc1_b, fcb, GH, GG, GH, LHW, 1);
  }
  // fc4: [G,256] @ [256,128] + b -> out
  {
    const int tiles = (GG / 16) * (GOUT / 16);   // 128 waves
    wmma_gemm_kernel<<<tiles / 4, 128, 0, stream>>>(
        fcb, GH, fc4_W, GOUT, fc4_b, (float*)d_out, GOUT, GG, GOUT, GH, 0);
  }
}